// FRGCN_20083267076628
// MI455X (gfx1250) — compile-verified
//
#include <hip/hip_runtime.h>

// ---------------------------------------------------------------------------
// FRGCN masked-min GCN propagation for MI455X (gfx1250, wave32).
//   6 iterations of: h = x@W (fp32 WMMA 16x16x4)  ->  scatter-min over edges
//   (order-preserving u32 atomic_min)  ->  elementwise combine.
// Memory-bound (~4 GB traffic total); GEMM done with V_WMMA_F32_16X16X4_F32
// to keep full fp32 precision while exercising the CDNA5 matrix pipe.
// W staged in LDS with K-pair interleaving so each B fragment is one
// aligned ds_load_b64 straight into an even VGPR pair (no mov packing).
// ---------------------------------------------------------------------------

typedef __attribute__((ext_vector_type(2))) float v2f;
typedef __attribute__((ext_vector_type(8))) float v8f;

#define DCH 64
#define CRIT 10000.0f

// Order-preserving map: float -> uint32 such that f0 < f1  <=>  enc(f0) < enc(f1).
__device__ __forceinline__ unsigned fenc(float f) {
    unsigned u = __float_as_uint(f);
    return (u & 0x80000000u) ? ~u : (u | 0x80000000u);
}
__device__ __forceinline__ float fdec(unsigned u) {
    unsigned b = (u & 0x80000000u) ? (u & 0x7FFFFFFFu) : ~u;
    return __uint_as_float(b);
}

// ---------------------------------------------------------------------------
// In-degree (graph-static, computed once per launch)
// ---------------------------------------------------------------------------
__global__ void deg_zero_kernel(int* __restrict__ deg, int N) {
    int i = blockIdx.x * blockDim.x + threadIdx.x;
    if (i < N) deg[i] = 0;
}

__global__ void deg_acc_kernel(const int* __restrict__ edge_index,
                               int* __restrict__ deg, int E) {
    int e = blockIdx.x * blockDim.x + threadIdx.x;
    if (e < E) {
        int d = edge_index[E + e];          // row 1 of [2,E] = dst
        atomicAdd(&deg[d], 1);
    }
}

// ---------------------------------------------------------------------------
// h = x @ W  via V_WMMA_F32_16X16X4_F32, and init agg buffer to enc(+inf).
// Block: 256 threads = 8 waves; each wave computes a 16x64 output strip.
// LDS layout: Wp[p][c] = (W[2p][c], W[2p+1][c]) as float2, p in [0,32).
// ---------------------------------------------------------------------------
__global__ __launch_bounds__(256) void gemm_init_kernel(
        const float* __restrict__ x, const float* __restrict__ W,
        float* __restrict__ h, unsigned* __restrict__ aggbits, int N) {
    __shared__ v2f Wp[32 * DCH];           // 16 KB, 8-byte aligned pairs
    int tid = threadIdx.x;
    // Stage W with K-pair interleave: element W[k][c] -> Wp[k/2][c].{x|y}
    float* Wps = (float*)Wp;
    for (int i = tid; i < DCH * DCH; i += 256) {
        int k = i >> 6, c = i & 63;
        Wps[((k >> 1) * DCH + c) * 2 + (k & 1)] = W[i];
    }
    __syncthreads();

    const int wave = tid >> 5;
    const int lane = tid & 31;
    const int hh   = lane >> 4;        // half-wave select (K pair / M+8)
    const int ml   = lane & 15;        // row (A) / col (B,C)
    const int row0 = blockIdx.x * 128 + wave * 16;

    // Clamp A row for loads (branchless, keeps EXEC all-ones for WMMA).
    int rA = row0 + ml;
    if (rA > N - 1) rA = N - 1;
    if (rA < 0) rA = 0;
    const float* __restrict__ xrow = x + (size_t)rA * DCH;

    v8f acc0 = {}, acc1 = {}, acc2 = {}, acc3 = {};

    #pragma unroll
    for (int kk = 0; kk < DCH; kk += 4) {
        const int kb = kk + 2 * hh;            // even K base for this half-wave
        // A frag: 16x4 fp32, lane -> row ml, K = kb, kb+1 (contiguous float2)
        v2f a = *(const v2f*)(xrow + kb);
        // B frags: one aligned 8-byte LDS load per 16-col tile
        const v2f* __restrict__ wp = &Wp[(kb >> 1) * DCH + ml];
        v2f b0 = wp[0];
        v2f b1 = wp[16];
        v2f b2 = wp[32];
        v2f b3 = wp[48];

        acc0 = __builtin_amdgcn_wmma_f32_16x16x4_f32(false, a, false, b0, (short)0, acc0, false, false);
        acc1 = __builtin_amdgcn_wmma_f32_16x16x4_f32(false, a, false, b1, (short)0, acc1, false, false);
        acc2 = __builtin_amdgcn_wmma_f32_16x16x4_f32(false, a, false, b2, (short)0, acc2, false, false);
        acc3 = __builtin_amdgcn_wmma_f32_16x16x4_f32(false, a, false, b3, (short)0, acc3, false, false);
    }

    // C/D layout: VGPR r -> M = row0 + r + 8*hh, col = ml (per 16-col tile)
    #pragma unroll
    for (int r = 0; r < 8; ++r) {
        int m = row0 + r + 8 * hh;
        if (m < N) {
            size_t base = (size_t)m * DCH + ml;
            h[base +  0] = acc0[r];
            h[base + 16] = acc1[r];
            h[base + 32] = acc2[r];
            h[base + 48] = acc3[r];
            aggbits[base +  0] = 0xFF800000u;   // enc(+inf)
            aggbits[base + 16] = 0xFF800000u;
            aggbits[base + 32] = 0xFF800000u;
            aggbits[base + 48] = 0xFF800000u;
        }
    }
}

// ---------------------------------------------------------------------------
// scatter-min: for each (edge, channel): aggbits[dst][c] = min(.., enc(h[src][c]*w))
// 64 threads per edge -> coalesced 256 B row transfers.
// ---------------------------------------------------------------------------
__global__ __launch_bounds__(256) void scatter_min_kernel(
        const float* __restrict__ h, const int* __restrict__ edge_index,
        const float* __restrict__ ew, unsigned* __restrict__ aggbits, int E) {
    long long idx = (long long)blockIdx.x * blockDim.x + threadIdx.x;
    int e  = (int)(idx >> 6);
    int ch = (int)(idx & 63);
    if (e < E) {
        int s = edge_index[e];
        int d = edge_index[E + e];
        float m = h[(size_t)s * DCH + ch] * ew[e];
        atomicMin(&aggbits[(size_t)d * DCH + ch], fenc(m));
    }
}

// ---------------------------------------------------------------------------
// combine: prop = (deg>0 ? dec(agg) : 0) + b ; phase1: keep-mask select,
// phase2: min-relaxation.
// ---------------------------------------------------------------------------
__global__ __launch_bounds__(256) void combine_kernel(
        const float* __restrict__ x, const unsigned* __restrict__ aggbits,
        const int* __restrict__ deg, const float* __restrict__ b,
        float* __restrict__ out, int N, int phase1) {
    long long i = (long long)blockIdx.x * blockDim.x + threadIdx.x;
    if (i < (long long)N * DCH) {
        int n  = (int)(i >> 6);
        int ch = (int)(i & 63);
        float agg  = (deg[n] > 0) ? fdec(aggbits[i]) : 0.0f;
        float prop = agg + b[ch];
        float xv   = x[i];
        out[i] = phase1 ? ((xv < CRIT) ? xv : prop) : fminf(xv, prop);
    }
}

// ---------------------------------------------------------------------------
// Host: 6 deterministic iterations, x double-buffered through workspace,
// final iteration writes d_out. Inputs never mutated.
// ---------------------------------------------------------------------------
extern "C" void kernel_launch(void* const* d_in, const int* in_sizes, int n_in,
                              void* d_out, int out_size, void* d_ws, size_t ws_size,
                              hipStream_t stream) {
    const float* x0 = (const float*)d_in[0];
    const int*   ei = (const int*)d_in[1];   // [2,E] (src row, dst row)
    const float* ew = (const float*)d_in[2];
    const float* W  = (const float*)d_in[3];
    const float* b  = (const float*)d_in[4];
    const int passes = 4;                    // fixed by setup_inputs()

    const int N = in_sizes[0] / DCH;
    const int E = in_sizes[2];

    char* ws = (char*)d_ws;
    size_t nd = (size_t)N * DCH * sizeof(float);
    float*    h       = (float*)(ws);
    unsigned* aggbits = (unsigned*)(ws + nd);
    float*    xA      = (float*)(ws + 2 * nd);
    float*    xB      = (float*)(ws + 3 * nd);
    int*      deg     = (int*)(ws + 4 * nd);

    // In-degree: graph-static, compute once per launch.
    deg_zero_kernel<<<(N + 255) / 256, 256, 0, stream>>>(deg, N);
    deg_acc_kernel<<<(E + 255) / 256, 256, 0, stream>>>(ei, deg, E);

    const int total = passes + passes / 2;   // 4 masked + 2 min-relax
    const float* cur = x0;
    const int gemm_blocks    = (N + 127) / 128;
    const int scatter_blocks = (int)(((long long)E * DCH + 255) / 256);
    const int comb_blocks    = (int)(((long long)N * DCH + 255) / 256);

    for (int it = 0; it < total; ++it) {
        float* out = (it == total - 1) ? (float*)d_out : ((it & 1) ? xB : xA);
        gemm_init_kernel<<<gemm_blocks, 256, 0, stream>>>(cur, W, h, aggbits, N);
        scatter_min_kernel<<<scatter_blocks, 256, 0, stream>>>(h, ei, ew, aggbits, E);
        combine_kernel<<<comb_blocks, 256, 0, stream>>>(cur, aggbits, deg, b, out, N,
                                                        (it < passes) ? 1 : 0);
        cur = out;
    }
}